// BruteForceRetrieval_78752520340100
// MI455X (gfx1250) — compile-verified
//
#include <hip/hip_runtime.h>

typedef __attribute__((ext_vector_type(2))) float v2f;
typedef __attribute__((ext_vector_type(8))) float v8f;
typedef __attribute__((ext_vector_type(4))) unsigned int v4u;
typedef __attribute__((ext_vector_type(4))) int v4i;
typedef __attribute__((ext_vector_type(8))) int v8i;

#define B_Q 512
#define N_CAND 500000
#define DIM 256
#define TOPK 100
#define CHUNK 2048
#define QT 32                                       // queries per block (2 M-tiles)
#define NUM_CHUNKS ((N_CAND + CHUNK - 1) / CHUNK)   // 245
#define QSTRIDE (DIM + 4)                           // pad -> conflict-free A-frag reads
#define NEG_INF (-3.402823466e+38f)

// ---------------------------------------------------------------------------
// Kernel 1: fused fp32 WMMA GEMM (32 queries x 2048 candidates per block)
//           + exact per-chunk sorted top-100 per query.
//           512 threads = 16 waves; ~289 KB LDS (CDNA5 320 KB WGP pool).
// ---------------------------------------------------------------------------
__global__ __launch_bounds__(512)
void score_chunk_topk(const float* __restrict__ queries,
                      const float* __restrict__ cand,
                      const int*   __restrict__ cand_ids,
                      float* __restrict__ pScore,
                      int*   __restrict__ pId)
{
    extern __shared__ float smem[];
    float* sQ = smem;                  // QT * QSTRIDE floats (32.5 KB, TDM-padded)
    float* sS = smem + QT * QSTRIDE;   // QT * CHUNK   floats (256 KB)

    const int tid   = threadIdx.x;
    const int lane  = tid & 31;
    const int wave  = tid >> 5;
    const int q0    = blockIdx.x * QT;
    const int chunk = blockIdx.y;
    const int cbase = chunk * CHUNK;

    // ---- Stage 32 query rows into LDS.
    // TDM path: one descriptor copies the contiguous 32x256 f32 tile and lets
    // the DMA engine insert the 4-dword row padding (pad_interval=256 dwords,
    // pad_amount=4 dwords) that realizes the bank-conflict-free QSTRIDE=260.
#if __has_builtin(__builtin_amdgcn_tensor_load_to_lds)
    if (wave == 0) {
        const unsigned long long ga =
            (unsigned long long)(uintptr_t)(queries + (size_t)q0 * DIM);
        const unsigned ldsa = (unsigned)(uintptr_t)sQ;   // low 32 bits = LDS offset
        v4u g0;
        g0[0] = 1u;                                  // count = 1 (valid user D#)
        g0[1] = ldsa;                                // lds_addr
        g0[2] = (unsigned)(ga & 0xffffffffu);        // global_addr[31:0]
        g0[3] = (unsigned)((ga >> 32) & 0x01ffffffu) // global_addr[56:32]
              | (2u << 30);                          // type = 2 ("image")
        v8i g1;
        g1[0] = (2 << 16)                            // data_size = 4 bytes
              | (1 << 20)                            // pad_enable
              | (7 << 22)                            // pad_interval = 256 dwords
              | (3 << 25);                           // pad_amount  = 4 dwords
        g1[1] = (int)(((QT * DIM) & 0xffff) << 16);  // tensor_dim0[15:0]  @ [63:48]
        g1[2] = (int)(((QT * DIM) >> 16) & 0xffff)   // tensor_dim0[31:16] @ [79:64]
              | (1 << 16);                           // tensor_dim1 = 1    @ [95:80]
        g1[3] = (int)(((QT * DIM) & 0xffff) << 16);  // tile_dim0 = 8192   @ [127:112]
        g1[4] = 1;                                   // tile_dim1 = 1; tile_dim2 = 0
        g1[5] = (int)(QT * DIM);                     // tensor_dim0_stride[31:0]
        g1[6] = 0;
        g1[7] = 0;
        v4i gz = {0, 0, 0, 0};
#if __clang_major__ >= 23
        v8i gz8 = {0, 0, 0, 0, 0, 0, 0, 0};
        __builtin_amdgcn_tensor_load_to_lds(g0, g1, gz, gz, gz8, 0);
#else
        __builtin_amdgcn_tensor_load_to_lds(g0, g1, gz, gz, 0);
#endif
        __builtin_amdgcn_s_wait_tensorcnt(0);
    }
#else
    for (int i = tid; i < QT * DIM; i += 512) {
        const int r = i >> 8;
        const int c = i & (DIM - 1);
        sQ[r * QSTRIDE + c] = queries[(q0 + r) * DIM + c];
    }
#endif
    __syncthreads();

    // WMMA fragment lane mapping (V_WMMA_F32_16X16X4_F32, wave32):
    //   A (16x4): lanes 0-15 -> M=lane, K=k..k+1 ; lanes 16-31 -> K=k+2..k+3
    //   B (4x16): lanes 0-15 -> N=lane, K=k..k+1 ; lanes 16-31 -> K=k+2..k+3
    //   C/D: VGPR r holds row M = r + 8*(lane>=16), column N = lane&15
    const int m  = lane & 15;
    const int kh = lane >> 4;

    // 128 column tiles of 16 candidates; wave w takes tiles w, w+16, ...
    // Each B fragment (one global_load_b64) feeds TWO WMMAs (2 M-tiles).
    for (int ct = wave; ct < CHUNK / 16; ct += 16) {
        const int l0    = ct * 16;
        const int nglob = cbase + l0 + m;                       // this lane's B column
        const int nsafe = nglob < N_CAND ? nglob : N_CAND - 1;
        const float* crow = cand + (size_t)nsafe * DIM;
        __builtin_prefetch(crow, 0, 0);                         // global_prefetch_b8

        v8f acc0 = {};
        v8f acc1 = {};
        #pragma unroll 4
        for (int k = 0; k < DIM; k += 4) {
            const int off = k + 2 * kh;
            v2f a0 = *(const v2f*)(sQ + m * QSTRIDE + off);            // ds_load_b64
            v2f a1 = *(const v2f*)(sQ + (m + 16) * QSTRIDE + off);     // ds_load_b64
            v2f b  = *(const v2f*)(crow + off);                        // global_load_b64
            acc0 = __builtin_amdgcn_wmma_f32_16x16x4_f32(
                false, a0, false, b, (short)0, acc0, false, false);
            acc1 = __builtin_amdgcn_wmma_f32_16x16x4_f32(
                false, a1, false, b, (short)0, acc1, false, false);
        }

        const bool valid = (nglob < N_CAND);
        #pragma unroll
        for (int r = 0; r < 8; ++r) {
            sS[(r + 8 * kh) * CHUNK + l0 + m]        = valid ? acc0[r] : NEG_INF;
            sS[(16 + r + 8 * kh) * CHUNK + l0 + m]   = valid ? acc1[r] : NEG_INF;
        }
    }
    __syncthreads();

    // Exact sorted top-100 of this chunk per query.
    // Wave w handles queries w and w+16 (2 queries/wave x 16 waves = 32).
    for (int qi = 0; qi < 2; ++qi) {
        const int q = wave + 16 * qi;
        float* row = sS + q * CHUNK;
        for (int it = 0; it < TOPK; ++it) {
            float best = NEG_INF;
            int   bidx = lane;
            for (int c = lane; c < CHUNK; c += 32) {
                const float v = row[c];
                if (v > best) { best = v; bidx = c; }
            }
            #pragma unroll
            for (int off2 = 16; off2 > 0; off2 >>= 1) {
                const float os = __shfl_down(best, off2, 32);
                const int   oi = __shfl_down(bidx, off2, 32);
                if (os > best || (os == best && oi < bidx)) { best = os; bidx = oi; }
            }
            bidx = __shfl(bidx, 0, 32);
            best = __shfl(best, 0, 32);
            if (lane == 0) {
                const size_t obase =
                    ((size_t)(q0 + q) * NUM_CHUNKS + chunk) * TOPK + it;
                pScore[obase] = best;
                const int ng = cbase + bidx;
                pId[obase]   = cand_ids[ng < N_CAND ? ng : N_CAND - 1];
                row[bidx]    = NEG_INF;   // same-wave DS ops are in-order
            }
        }
    }
}

// ---------------------------------------------------------------------------
// Kernel 2: one block per query; 245-way merge of sorted top-100 lists
//           via repeated argmax over list heads. Exact global top-100.
// ---------------------------------------------------------------------------
__global__ __launch_bounds__(256)
void merge_topk(const float* __restrict__ pScore,
                const int*   __restrict__ pId,
                float* __restrict__ outScore,
                int*   __restrict__ outId)
{
    __shared__ float sHead[256];
    __shared__ int   sPtr[256];
    __shared__ float sRedS[256];
    __shared__ int   sRedI[256];

    const int tid = threadIdx.x;
    const int q   = blockIdx.x;
    const size_t base = (size_t)q * NUM_CHUNKS * TOPK;

    sPtr[tid]  = 0;
    sHead[tid] = (tid < NUM_CHUNKS) ? pScore[base + (size_t)tid * TOPK] : NEG_INF;
    __syncthreads();

    for (int it = 0; it < TOPK; ++it) {
        sRedS[tid] = sHead[tid];
        sRedI[tid] = tid;
        __syncthreads();
        #pragma unroll
        for (int s = 128; s > 0; s >>= 1) {
            if (tid < s) {
                if (sRedS[tid + s] > sRedS[tid] ||
                    (sRedS[tid + s] == sRedS[tid] && sRedI[tid + s] < sRedI[tid])) {
                    sRedS[tid] = sRedS[tid + s];
                    sRedI[tid] = sRedI[tid + s];
                }
            }
            __syncthreads();
        }
        if (tid == 0) {
            const int w = sRedI[0];
            int p = sPtr[w];
            outScore[(size_t)q * TOPK + it] = sRedS[0];
            outId  [(size_t)q * TOPK + it] = pId[base + (size_t)w * TOPK + p];
            ++p;
            sPtr[w]  = p;
            sHead[w] = (p < TOPK) ? pScore[base + (size_t)w * TOPK + p] : NEG_INF;
        }
        __syncthreads();
    }
}

// ---------------------------------------------------------------------------
extern "C" void kernel_launch(void* const* d_in, const int* in_sizes, int n_in,
                              void* d_out, int out_size, void* d_ws, size_t ws_size,
                              hipStream_t stream)
{
    const float* queries  = (const float*)d_in[0];   // [512, 256] f32
    const float* cand     = (const float*)d_in[1];   // [500000, 256] f32
    const int*   cand_ids = (const int*)d_in[2];     // [500000] i32
    // d_in[3] = k (==100), fixed at compile time.

    // Workspace: per-(query, chunk) sorted top-100 (score f32 + id i32) ~ 100 MB.
    float* pScore = (float*)d_ws;
    int*   pId    = (int*)((char*)d_ws +
                           (size_t)B_Q * NUM_CHUNKS * TOPK * sizeof(float));

    float* outScore = (float*)d_out;                                   // [512,100]
    int*   outId    = (int*)((float*)d_out + (size_t)B_Q * TOPK);      // [512,100]

    const dim3 grid1(B_Q / QT, NUM_CHUNKS);                // 16 x 245
    const size_t lds1 = (size_t)(QT * QSTRIDE + QT * CHUNK) * sizeof(float); // ~289 KB
    hipLaunchKernelGGL(score_chunk_topk, grid1, dim3(512), lds1, stream,
                       queries, cand, cand_ids, pScore, pId);

    hipLaunchKernelGGL(merge_topk, dim3(B_Q), dim3(256), 0, stream,
                       pScore, pId, outScore, outId);
}